// SharedHiddenStateSpatiotemporalModel_3779571220683
// MI455X (gfx1250) — compile-verified
//
#include <hip/hip_runtime.h>
#include <hip/hip_bf16.h>
#include <cstdint>

typedef __attribute__((ext_vector_type(2))) float v2f;
typedef __attribute__((ext_vector_type(8))) float v8f;

#define NN 256
#define TT 1024
#define KK 1000
#define DD 64
#define EPSF 1e-8f

// ---------------------------------------------------------------------------
// Kernel 1: type_emb[row, :] = type_table[event_types[row], :]
// 64 MB of coalesced float4 (b128) stores; table (256 KB) is L2-resident.
// One float4 per thread; 16 threads per row.
// ---------------------------------------------------------------------------
__global__ void emb_gather_kernel(const int* __restrict__ ev,
                                  const float* __restrict__ table,
                                  float* __restrict__ emb_out) {
    int idx = blockIdx.x * blockDim.x + threadIdx.x;   // over N*T*16 float4s
    int row = idx >> 4;
    int q   = idx & 15;
    int et  = ev[row];
    const float4* src = reinterpret_cast<const float4*>(table);
    float4 v = src[et * 16 + q];
    reinterpret_cast<float4*>(emb_out)[idx] = v;       // idx == row*16 + q
}

// ---------------------------------------------------------------------------
// Kernel 2: per-trajectory reductions via V_WMMA_F32_16X16X4_F32 with B=ones.
//   D = A x Ones accumulates row sums of A tiles in f32.
// Block = 256 threads (8 waves), handles 16 trajectories (rows).
// Wave w reduces columns [w*128, (w+1)*128) with 32 chained WMMAs per term.
// A-layout (32-bit 16x4): lanes 0-15 hold K=0,1 ; lanes 16-31 hold K=2,3.
// D-layout: lanes 0-15 VGPR r = D[r, n] ; lanes 16-31 VGPR r = D[8+r, n-16].
// All columns of D are equal (B is all ones), so lane 0 / lane 16 carry the
// 16 per-row slab sums; waves are combined through LDS (fixed order).
// ---------------------------------------------------------------------------
__global__ void traj_reduce_kernel(const int*   __restrict__ ev,
                                   const float* __restrict__ prob,
                                   const float* __restrict__ inten,
                                   const float* __restrict__ Lam,
                                   const float* __restrict__ mask,
                                   float* __restrict__ ll_out,
                                   float* __restrict__ tnt_out) {
    const int tid     = threadIdx.x;
    const int lane    = tid & 31;
    const int wave    = tid >> 5;
    const int rowBase = blockIdx.x * 16;
    const int row     = rowBase + (lane & 15);
    const int khalf   = (lane >> 4) * 2;       // 0 for lanes 0-15, 2 for 16-31
    const long rowOff = (long)row * TT;

    v8f ct = {};   // accumulator: time term row sums
    v8f ce = {};   // accumulator: event term row sums
    v2f bones; bones.x = 1.0f; bones.y = 1.0f;

    const int cbase = wave * 128;
    for (int t = 0; t < 32; ++t) {
        const int c0 = cbase + 4 * t + khalf;
        v2f at, ae;
#pragma unroll
        for (int u = 0; u < 2; ++u) {
            const int  c  = c0 + u;
            const long o  = rowOff + c;
            const float m = mask[o];
            const float tv = __logf(inten[o] + EPSF) * m;
            const int   e  = ev[o];
            const float p  = prob[o * KK + e];
            const float evv = __logf(p + EPSF) * m;
            if (u == 0) { at.x = tv; ae.x = evv; }
            else        { at.y = tv; ae.y = evv; }
        }
        // (neg_a, A, neg_b, B, c_mod, C, reuse_a, reuse_b)
        ct = __builtin_amdgcn_wmma_f32_16x16x4_f32(false, at, false, bones,
                                                   (short)0, ct, false, false);
        ce = __builtin_amdgcn_wmma_f32_16x16x4_f32(false, ae, false, bones,
                                                   (short)0, ce, false, false);
    }

    __shared__ float st[8][16];
    __shared__ float se[8][16];
    if (lane == 0) {
#pragma unroll
        for (int r = 0; r < 8; ++r) { st[wave][r] = ct[r]; se[wave][r] = ce[r]; }
    }
    if (lane == 16) {
#pragma unroll
        for (int r = 0; r < 8; ++r) { st[wave][8 + r] = ct[r]; se[wave][8 + r] = ce[r]; }
    }
    __syncthreads();

    if (tid < 16) {
        float s_t = 0.0f, s_e = 0.0f;
#pragma unroll
        for (int w = 0; w < 8; ++w) { s_t += st[w][tid]; s_e += se[w][tid]; }
        const int rg = rowBase + tid;
        const float tnt = s_t - Lam[rg];
        tnt_out[rg] = tnt;
        ll_out[rg]  = tnt + s_e;
    }
}

// ---------------------------------------------------------------------------
// Kernel 3: deterministic tree reduction of ll_N (256 values) -> scalar.
// ---------------------------------------------------------------------------
__global__ void final_sum_kernel(const float* __restrict__ ll,
                                 float* __restrict__ out_scalar) {
    __shared__ float buf[256];
    const int tid = threadIdx.x;
    buf[tid] = ll[tid];
    __syncthreads();
    for (int s = 128; s > 0; s >>= 1) {
        if (tid < s) buf[tid] += buf[tid + s];
        __syncthreads();
    }
    if (tid == 0) *out_scalar = buf[0];
}

// ---------------------------------------------------------------------------
extern "C" void kernel_launch(void* const* d_in, const int* in_sizes, int n_in,
                              void* d_out, int out_size, void* d_ws, size_t ws_size,
                              hipStream_t stream) {
    const int*   ev    = (const int*)  d_in[0];   // (N,T)   int32
    const float* prob  = (const float*)d_in[1];   // (N,T,K) f32
    const float* inten = (const float*)d_in[2];   // (N,T)   f32
    const float* Lam   = (const float*)d_in[3];   // (N,)    f32
    const float* mask  = (const float*)d_in[4];   // (N,T)   f32
    const float* table = (const float*)d_in[5];   // (K,D)   f32

    float* out    = (float*)d_out;
    float* ll     = out;            // [0,256)      ll_N
    float* scalar = out + 256;      // [256]        time_loglik
    float* tnt    = out + 257;      // [257,513)    time_loglik_NT
    float* emb    = out + 513;      // [513, ...)   type_emb (N*T*D)

    // 64 MB coalesced embedding gather: N*T rows, 16 float4 lanes per row.
    emb_gather_kernel<<<(NN * TT * 16) / 256, 256, 0, stream>>>(ev, table, emb);

    // 16 blocks x 256 threads: 16 trajectories per block, WMMA row-sum chains.
    traj_reduce_kernel<<<NN / 16, 256, 0, stream>>>(ev, prob, inten, Lam, mask,
                                                    ll, tnt);

    // Scalar total.
    final_sum_kernel<<<1, 256, 0, stream>>>(ll, scalar);
}